// GNN_18588618457604
// MI455X (gfx1250) — compile-verified
//
#include <hip/hip_runtime.h>

typedef __attribute__((ext_vector_type(2))) float v2f;
typedef __attribute__((ext_vector_type(8))) float v8f;

#define D 512          // d_in == d_hid == d_out == 512 in the reference
#define TILE_M 16

// -----------------------------------------------------------------------------
// GEMM: H[m, n] = sum_k Xin[m, k] * W[n, k] + bias[n]   (i.e. X @ W^T + b)
// Optional ReLU applied to Xin while staging into LDS (for layer 2's input).
// One block = one 16-row M tile; 8 waves x 64 columns = full 512-wide output.
// Each wave: 4 accumulators (16x16 each) sharing one A fragment per k-step.
// -----------------------------------------------------------------------------
__global__ __launch_bounds__(256) void gemm_wmma_f32(
    const float* __restrict__ Xin, const float* __restrict__ W,
    const float* __restrict__ bias, float* __restrict__ H, int relu_in)
{
    __shared__ float xs[TILE_M * D];          // 16 x 512 fp32 = 32 KB
    const int m16 = blockIdx.x * TILE_M;
    const int t   = threadIdx.x;

    // Cooperative load of the 16x512 X tile (optionally ReLU'ed) into LDS.
    const float4* xin4 = (const float4*)(Xin + (size_t)m16 * D);
    float4* xs4 = (float4*)xs;
#pragma unroll
    for (int i = 0; i < 8; ++i) {
        int idx = t + i * 256;                // float4 index within tile
        float4 v = xin4[idx];
        if (relu_in) {
            v.x = fmaxf(v.x, 0.0f); v.y = fmaxf(v.y, 0.0f);
            v.z = fmaxf(v.z, 0.0f); v.w = fmaxf(v.w, 0.0f);
        }
        xs4[idx] = v;
    }
    __syncthreads();

    const int wave  = t >> 5;                 // 0..7
    const int lane  = t & 31;
    const int row   = lane & 15;              // M (A frag) / N-row-of-W (B frag)
    const int koff  = (lane >> 4) << 1;       // 0 or 2 : K base for this half-wave
    const int nbase = wave * 64;

    v8f acc0 = {}, acc1 = {}, acc2 = {}, acc3 = {};

    const float* ar  = xs + row * D;
    const float* wr0 = W + (size_t)(nbase +  0 + row) * D;
    const float* wr1 = W + (size_t)(nbase + 16 + row) * D;
    const float* wr2 = W + (size_t)(nbase + 32 + row) * D;
    const float* wr3 = W + (size_t)(nbase + 48 + row) * D;

#pragma unroll 2
    for (int k = 0; k < D; k += 4) {
        const int kk = k + koff;
        // A fragment from LDS: lane (row, K=kk..kk+1)  -> ds_load_b64
        v2f a  = *(const v2f*)(ar  + kk);
        // B fragments from global (L2-resident W): W[n+row, kk..kk+1]
        v2f b0 = *(const v2f*)(wr0 + kk);
        v2f b1 = *(const v2f*)(wr1 + kk);
        v2f b2 = *(const v2f*)(wr2 + kk);
        v2f b3 = *(const v2f*)(wr3 + kk);
        acc0 = __builtin_amdgcn_wmma_f32_16x16x4_f32(false, a, false, b0, (short)0, acc0, false, false);
        acc1 = __builtin_amdgcn_wmma_f32_16x16x4_f32(false, a, false, b1, (short)0, acc1, false, false);
        acc2 = __builtin_amdgcn_wmma_f32_16x16x4_f32(false, a, false, b2, (short)0, acc2, false, false);
        acc3 = __builtin_amdgcn_wmma_f32_16x16x4_f32(false, a, false, b3, (short)0, acc3, false, false);
    }

    // C/D layout: vgpr v holds row m16 + v + 8*(lane>>4), col = nsub + (lane&15).
    const int rbase = m16 + ((lane >> 4) << 3);
    const int c0 = nbase + (lane & 15);
#pragma unroll
    for (int s = 0; s < 4; ++s) {
        const v8f acc = (s == 0) ? acc0 : (s == 1) ? acc1 : (s == 2) ? acc2 : acc3;
        const int col = c0 + s * 16;
        const float bv = bias[col];
        float* hp = H + (size_t)rbase * D + col;
#pragma unroll
        for (int v = 0; v < 8; ++v) {
            hp[(size_t)v * D] = acc[v] + bv;
        }
    }
}

// -----------------------------------------------------------------------------
// Zero-fill (float4 stores).
// -----------------------------------------------------------------------------
__global__ __launch_bounds__(256) void zero_f32(float* __restrict__ p, int n4)
{
    int i = blockIdx.x * 256 + threadIdx.x;
    if (i < n4) ((float4*)p)[i] = make_float4(0.f, 0.f, 0.f, 0.f);
}

// -----------------------------------------------------------------------------
// Edge scatter-add: agg[dst[e], :] += h[src[e], :]
// 128 threads per edge, float4 gather + 4 global_atomic_add_f32 each.
// Destination rows (20.5 MB) are fully L2-resident -> atomics stay on-chip.
// -----------------------------------------------------------------------------
__global__ __launch_bounds__(256) void scatter_add(
    const float* __restrict__ H, const int* __restrict__ src,
    const int* __restrict__ dst, float* __restrict__ AGG, int E)
{
    long long tid = (long long)blockIdx.x * 256 + threadIdx.x;
    int e = (int)(tid >> 7);
    if (e >= E) return;
    int c = (int)(tid & 127) << 2;            // column base, coalesced per edge
    int s = src[e];
    int d = dst[e];
    float4 v = *(const float4*)(H + (size_t)s * D + c);
    float* out = AGG + (size_t)d * D + c;
    atomicAdd(out + 0, v.x);
    atomicAdd(out + 1, v.y);
    atomicAdd(out + 2, v.z);
    atomicAdd(out + 3, v.w);
}

// -----------------------------------------------------------------------------
// In-place ReLU (final output).
// -----------------------------------------------------------------------------
__global__ __launch_bounds__(256) void relu_f32(float* __restrict__ p, int n4)
{
    int i = blockIdx.x * 256 + threadIdx.x;
    if (i < n4) {
        float4 v = ((float4*)p)[i];
        v.x = fmaxf(v.x, 0.0f); v.y = fmaxf(v.y, 0.0f);
        v.z = fmaxf(v.z, 0.0f); v.w = fmaxf(v.w, 0.0f);
        ((float4*)p)[i] = v;
    }
}

// -----------------------------------------------------------------------------
// Launch: gemm1 -> zero -> scatter1 -> gemm2(relu-in fused) -> zero -> scatter2
//         -> relu. All on `stream`; scratch = 2 * N*D floats in d_ws.
// -----------------------------------------------------------------------------
extern "C" void kernel_launch(void* const* d_in, const int* in_sizes, int n_in,
                              void* d_out, int out_size, void* d_ws, size_t ws_size,
                              hipStream_t stream)
{
    const float* x  = (const float*)d_in[0];
    const int*   ei = (const int*)d_in[1];
    const float* W1 = (const float*)d_in[2];
    const float* b1 = (const float*)d_in[3];
    const float* W2 = (const float*)d_in[4];
    const float* b2 = (const float*)d_in[5];
    float* out = (float*)d_out;

    const int N = in_sizes[0] / D;            // 10000 nodes
    const int E = in_sizes[1] / 2;            // 160000 edges
    const int* src = ei;
    const int* dst = ei + E;

    float* h   = (float*)d_ws;                // N*D fp32
    float* agg = h + (size_t)N * D;           // N*D fp32

    const int nd   = N * D;
    const int n4   = nd / 4;
    const int mtb  = N / TILE_M;              // 625 M tiles
    const int zb   = (n4 + 255) / 256;
    const int sb   = (int)(((long long)E * 128 + 255) / 256);

    // ---- Layer 1 ----
    gemm_wmma_f32<<<mtb, 256, 0, stream>>>(x, W1, b1, h, 0);
    zero_f32<<<zb, 256, 0, stream>>>(agg, n4);
    scatter_add<<<sb, 256, 0, stream>>>(h, src, dst, agg, E);
    // relu(agg) fused into gemm2's LDS staging

    // ---- Layer 2 ----
    gemm_wmma_f32<<<mtb, 256, 0, stream>>>(agg, W2, b2, h, 1);
    zero_f32<<<zb, 256, 0, stream>>>(out, n4);
    scatter_add<<<sb, 256, 0, stream>>>(h, src, dst, out, E);
    relu_f32<<<zb, 256, 0, stream>>>(out, n4);
}